// TopK_17136919511768
// MI455X (gfx1250) — compile-verified
//
#include <hip/hip_runtime.h>
#include <stdint.h>

#define TOPK 8
#define NGROUP 8
#define TOPK_GROUP 4
#define NEXPERT 256
#define SCALING 2.5f
#define WAVES_PER_BLOCK 8

// Monotone float -> uint32 key: a > b  <=>  fkey(a) > fkey(b)
__device__ __forceinline__ uint32_t fkey(float f) {
    uint32_t u = __float_as_uint(f);
    return (u & 0x80000000u) ? ~u : (u | 0x80000000u);
}

// Async-copy one 1KB token row (256 fp32) global -> LDS using the CDNA5
// ASYNCcnt-tracked path. 32 lanes x 16B x 2 issues = 1024B.
__device__ __forceinline__ void issue_row(const float* gsrc, const float* lds_dst, int lane) {
    uint64_t ga = (uint64_t)(uintptr_t)gsrc + (uint32_t)(lane * 16);
    // generic shared pointer: low 32 bits are the LDS byte offset (ISA 10.2 aperture calc)
    uint32_t la = (uint32_t)(uintptr_t)lds_dst + (uint32_t)(lane * 16);
    asm volatile(
        "global_load_async_to_lds_b128 %0, %2, off\n\t"
        "global_load_async_to_lds_b128 %1, %3, off"
        :: "v"(la), "v"(la + 512u), "v"(ga), "v"(ga + 512ull)
        : "memory");
}

__global__ __launch_bounds__(256) void deepseek_route_kernel(
    const float* __restrict__ logits,     // [T, 256]
    const float* __restrict__ bias,       // [256]
    float* __restrict__ w_out,            // [T, 8]
    float* __restrict__ id_out,           // [T, 8] (ids stored as float)
    int num_token)
{
    __shared__ float smem[WAVES_PER_BLOCK][2][NEXPERT];   // 16 KB/block, double buffered per wave
    const int lane  = threadIdx.x & 31;
    const int wid   = threadIdx.x >> 5;
    const int gwave = blockIdx.x * WAVES_PER_BLOCK + wid;
    const int nwave = gridDim.x * WAVES_PER_BLOCK;

    // Per-lane correction bias for experts 8*lane .. 8*lane+7 (tiny, L2-resident)
    float b[8];
    {
        const float4* bp = reinterpret_cast<const float4*>(bias) + lane * 2;
        float4 b0 = bp[0], b1 = bp[1];
        b[0]=b0.x; b[1]=b0.y; b[2]=b0.z; b[3]=b0.w;
        b[4]=b1.x; b[5]=b1.y; b[6]=b1.z; b[7]=b1.w;
    }

    if (gwave < num_token) issue_row(logits + (size_t)gwave * NEXPERT, &smem[wid][0][0], lane);

    int it = 0;
    for (int tok = gwave; tok < num_token; tok += nwave, ++it) {
        const int nxt = tok + nwave;
        // All prior DS reads of the buffer we are about to overwrite must retire:
        // async LDS writes are NOT ordered against in-flight ds_loads (ISA 4.1).
        asm volatile("s_wait_dscnt 0x0" ::: "memory");
        if (nxt < num_token) {
            issue_row(logits + (size_t)nxt * NEXPERT, &smem[wid][(it + 1) & 1][0], lane);
            asm volatile("s_wait_asynccnt 0x2" ::: "memory");  // current buffer's 2 ops done
        } else {
            asm volatile("s_wait_asynccnt 0x0" ::: "memory");
        }
        const float* row = &smem[wid][it & 1][0];

        // ---- load my 8 unbiased logits from LDS, form biased scores ----
        float4 x0 = *reinterpret_cast<const float4*>(row + lane * 8);
        float4 x1 = *reinterpret_cast<const float4*>(row + lane * 8 + 4);
        float x[8] = {x0.x, x0.y, x0.z, x0.w, x1.x, x1.y, x1.z, x1.w};
        float s[8];
        #pragma unroll
        for (int j = 0; j < 8; ++j) s[j] = x[j] + b[j];

        // ---- group score: sum of top-2 biased scores in each 32-expert group ----
        float m1 = -INFINITY, m2 = -INFINITY;           // local top-2 of my 8
        #pragma unroll
        for (int j = 0; j < 8; ++j) {
            float hi = fmaxf(m1, s[j]);
            float lo = fminf(m1, s[j]);
            m2 = fmaxf(m2, lo);
            m1 = hi;
        }
        #pragma unroll
        for (int k = 1; k <= 2; k <<= 1) {              // merge across the 4 lanes of my group
            float o1 = __shfl_xor(m1, k, 32);
            float o2 = __shfl_xor(m2, k, 32);
            float hi = fmaxf(m1, o1), lo = fminf(m1, o1);
            m2 = fmaxf(fmaxf(m2, o2), lo);
            m1 = hi;
        }
        const float gscore = m1 + m2;

        // ---- rank my group among all 8; keep top-4 (ties -> lower group index) ----
        const int myg = lane >> 2;
        int rank = 0;
        #pragma unroll
        for (int g = 0; g < NGROUP; ++g) {
            float og = __shfl(gscore, g * 4, 32);
            rank += (og > gscore) || ((og == gscore) && (g < myg));
        }
        const bool keep = (rank < TOPK_GROUP);

        // ---- exact top-8 of masked biased scores (jax tie-break: lowest index) ----
        uint32_t kk[8];
        #pragma unroll
        for (int j = 0; j < 8; ++j) kk[j] = keep ? fkey(s[j]) : 0u;  // 0 < fkey(any finite)

        float sum = 0.0f, myw = 0.0f;
        int myid = 0;
        for (int t = 0; t < TOPK; ++t) {
            // local argmax, smallest j wins ties (strict >)
            uint32_t bk = kk[0]; int bj = 0;
            #pragma unroll
            for (int j = 1; j < 8; ++j) {
                bool c = kk[j] > bk;
                bk = c ? kk[j] : bk;
                bj = c ? j : bj;
            }
            // wave-wide max of the 32-bit key
            uint32_t wk = bk;
            #pragma unroll
            for (int m = 16; m >= 1; m >>= 1) {
                uint32_t o = __shfl_xor(wk, m, 32);
                wk = (o > wk) ? o : wk;
            }
            // lowest lane holding the max == lowest expert index (lane order is ascending)
            unsigned long long mb = __ballot(bk == wk);
            int wl = __ffsll(mb) - 1;
            int wj = __shfl(bj, wl, 32);
            int widx = wl * 8 + wj;
            float wval = row[widx];          // UNbiased logit, broadcast LDS read
            sum += wval;
            if (lane == t) { myw = wval; myid = widx; }
            #pragma unroll
            for (int j = 0; j < 8; ++j)      // remove winner from its lane
                if (lane == wl && j == wj) kk[j] = 0u;
        }

        const float scale = SCALING / sum;
        if (lane < TOPK) {
            w_out[(size_t)tok * TOPK + lane]  = myw * scale;
            id_out[(size_t)tok * TOPK + lane] = (float)myid;
        }
    }
}

extern "C" void kernel_launch(void* const* d_in, const int* in_sizes, int n_in,
                              void* d_out, int out_size, void* d_ws, size_t ws_size,
                              hipStream_t stream) {
    const float* logits = (const float*)d_in[0];
    const float* bias   = (const float*)d_in[1];
    const int num_token = in_sizes[0] / NEXPERT;

    float* w_out  = (float*)d_out;
    float* id_out = w_out + (size_t)num_token * TOPK;

    int total_waves_wanted = (num_token + 7) / 8;        // ~8 tokens per wave
    int blocks = (total_waves_wanted + WAVES_PER_BLOCK - 1) / WAVES_PER_BLOCK;
    if (blocks > 2048) blocks = 2048;
    if (blocks < 1) blocks = 1;

    deepseek_route_kernel<<<blocks, WAVES_PER_BLOCK * 32, 0, stream>>>(
        logits, bias, w_out, id_out, num_token);
}